// CustomAttentionLayer_25271587570312
// MI455X (gfx1250) — compile-verified
//
#include <hip/hip_runtime.h>
#include <hip/hip_bf16.h>

typedef __attribute__((ext_vector_type(2))) float v2f;
typedef __attribute__((ext_vector_type(8))) float v8f;

#define D_FEAT 128

// Monotonic unsigned encoding of float for atomicMax-based segment max.
__device__ __forceinline__ unsigned enc_f32(float f) {
  unsigned u = __float_as_uint(f);
  return (u & 0x80000000u) ? ~u : (u | 0x80000000u);
}
__device__ __forceinline__ float dec_f32(unsigned u) {
  return (u & 0x80000000u) ? __uint_as_float(u & 0x7FFFFFFFu) : __uint_as_float(~u);
}

// D = rowscale(A) @ B^T + bias   (A: MxK row-major, Bt: NxK row-major, D: MxN)
// One wave per 16x16 output tile, K accumulated via V_WMMA_F32_16X16X4_F32.
// Inner loop is pure load+wmma; when SCALED, the per-row softmax normalization
// 1/(denom[m]+1e-16) is applied in the epilogue (row-scale commutes with GEMM).
// Requires M % 16 == 0 (holds: 10000 = 625*16), K % 4 == 0, N % 16 == 0.
template <bool SCALED>
__global__ void gemm_nt_wmma(const float* __restrict__ A, const float* __restrict__ Bt,
                             const float* __restrict__ bias,
                             const float* __restrict__ denom,
                             float* __restrict__ D, int M, int K, int N) {
  int wave = (blockIdx.x * blockDim.x + threadIdx.x) >> 5;
  int lane = threadIdx.x & 31;
  int ntiles = N >> 4;
  int mt = wave / ntiles;
  int nt = wave - mt * ntiles;
  if (mt * 16 >= M) return;                 // wave-uniform: EXEC stays all-1s for WMMA
  int half = lane >> 4;                     // 0: lanes 0-15, 1: lanes 16-31
  int l16  = lane & 15;
  int m = mt * 16 + l16;                    // A row held by this lane
  int n = nt * 16 + l16;                    // Bt row (= output column) held by this lane
  int kb = half * 2;                        // K sub-offset within each K=4 step

  const float* arow = A  + (size_t)m * K;
  const float* brow = Bt + (size_t)n * K;

  v8f acc = {};
  #pragma unroll 4
  for (int k0 = 0; k0 < K; k0 += 4) {
    v2f a = *(const v2f*)(arow + k0 + kb);  // A[m][k0+kb .. +1]
    v2f b = *(const v2f*)(brow + k0 + kb);  // B[k][n] = Bt row n, contiguous
    acc = __builtin_amdgcn_wmma_f32_16x16x4_f32(false, a, false, b,
                                                (short)0, acc, false, false);
  }

  float bn = bias[n];
  int row0 = mt * 16 + half * 8;            // C/D layout: VGPR j -> row j + 8*half, col l16
  int colo = nt * 16 + l16;
  #pragma unroll
  for (int j = 0; j < 8; ++j) {
    float v = acc[j];
    if (SCALED) v = v / (denom[row0 + j] + 1e-16f);   // wave-broadcast load, tiny array
    D[(size_t)(row0 + j) * N + colo] = v + bn;
  }
}

// g_node[i] = dot(x[i,:], W_gate[0,:]) + b_gate   — one wave per node, float4 per lane.
__global__ void gate_kernel(const float* __restrict__ x, const float* __restrict__ wg,
                            const float* __restrict__ bg, float* __restrict__ g_node,
                            int n_nodes) {
  int wave = (blockIdx.x * blockDim.x + threadIdx.x) >> 5;
  int lane = threadIdx.x & 31;
  if (wave >= n_nodes) return;
  float4 a = ((const float4*)(x + (size_t)wave * D_FEAT))[lane];
  float4 b = ((const float4*)wg)[lane];
  float s = a.x * b.x + a.y * b.y + a.z * b.z + a.w * b.w;
  #pragma unroll
  for (int off = 16; off > 0; off >>= 1) s += __shfl_xor(s, off, 32);
  if (lane == 0) g_node[wave] = s + bg[0];
}

// Pass 1 over edges: segment max of gathered gate values (ordered-uint atomicMax).
__global__ void edge_max_kernel(const int* __restrict__ row_idx, const int* __restrict__ col_idx,
                                const float* __restrict__ g_node, unsigned* __restrict__ segmax,
                                int n_edges) {
  int e = blockIdx.x * blockDim.x + threadIdx.x;
  if (e >= n_edges) return;
  int r = row_idx[e], c = col_idx[e];
  atomicMax(&segmax[r], enc_f32(g_node[c]));
}

// Pass 2 over edges: e = exp(g[col]-max[row]); denom[row]+=e; pooled[row,:]+=e*y[col,:].
// One wave per edge; each lane handles a float4 chunk (32*4 = 128 features).
__global__ void edge_acc_kernel(const int* __restrict__ row_idx, const int* __restrict__ col_idx,
                                const float* __restrict__ g_node, const unsigned* __restrict__ segmax,
                                const float* __restrict__ y,
                                float* __restrict__ denom, float* __restrict__ pooled,
                                int n_edges) {
  int wave = (blockIdx.x * blockDim.x + threadIdx.x) >> 5;
  int lane = threadIdx.x & 31;
  if (wave >= n_edges) return;
  int r = row_idx[wave], c = col_idx[wave];        // same-address loads -> broadcast
  float e = __expf(g_node[c] - dec_f32(segmax[r]));
  if (lane == 0) atomicAdd(&denom[r], e);
  float4 v = ((const float4*)(y + (size_t)c * D_FEAT))[lane];
  float* pr = pooled + (size_t)r * D_FEAT + lane * 4;
  atomicAdd(pr + 0, e * v.x);
  atomicAdd(pr + 1, e * v.y);
  atomicAdd(pr + 2, e * v.z);
  atomicAdd(pr + 3, e * v.w);
}

extern "C" void kernel_launch(void* const* d_in, const int* in_sizes, int n_in,
                              void* d_out, int out_size, void* d_ws, size_t ws_size,
                              hipStream_t stream) {
  const float* x      = (const float*)d_in[0];
  const int*   eidx   = (const int*)d_in[1];   // edge_index, (2, E) flattened
  const float* W_lin  = (const float*)d_in[3];
  const float* b_lin  = (const float*)d_in[4];
  const float* W_gate = (const float*)d_in[5];
  const float* b_gate = (const float*)d_in[6];
  const float* W_out  = (const float*)d_in[7];
  const float* b_out  = (const float*)d_in[8];
  float* out = (float*)d_out;

  const int n_nodes = in_sizes[0] / D_FEAT;
  const int n_edges = in_sizes[1] / 2;
  const int* row_idx = eidx;
  const int* col_idx = eidx + n_edges;

  // Workspace layout (all L2-resident, ~10.4 MB):
  //   [pooled: N*128 f32][denom: N f32][segmax: N u32][y: N*128 f32][g_node: N f32]
  size_t NM = (size_t)n_nodes * D_FEAT;
  float*    pooled = (float*)d_ws;
  float*    denom  = pooled + NM;
  unsigned* segmax = (unsigned*)(denom + n_nodes);
  float*    y      = (float*)(segmax + n_nodes);
  float*    g_node = y + NM;

  // Zero pooled+denom+segmax in one stream-ordered memset (graph-capturable).
  // segmax=0 is a valid -inf substitute: every finite encoded gate value > 0.
  hipMemsetAsync(d_ws, 0, (NM + 2 * (size_t)n_nodes) * sizeof(float), stream);

  const int mtiles = (n_nodes + 15) / 16;
  const int ntiles = D_FEAT / 16;
  const int gemm_threads = mtiles * ntiles * 32;   // one wave per 16x16 tile

  // y = x @ W_lin^T + b_lin   (WMMA f32, exact, pure load+wmma inner loop)
  gemm_nt_wmma<false><<<(gemm_threads + 255) / 256, 256, 0, stream>>>(
      x, W_lin, b_lin, nullptr, y, n_nodes, D_FEAT, D_FEAT);

  // per-node gate scalar
  gate_kernel<<<(n_nodes * 32 + 255) / 256, 256, 0, stream>>>(
      x, W_gate, b_gate, g_node, n_nodes);

  // edge pass 1: segment max
  edge_max_kernel<<<(n_edges + 255) / 256, 256, 0, stream>>>(
      row_idx, col_idx, g_node, segmax, n_edges);

  // edge pass 2: un-normalized softmax accumulate (wave per edge)
  long long acc_threads = (long long)n_edges * 32;
  edge_acc_kernel<<<(unsigned)((acc_threads + 255) / 256), 256, 0, stream>>>(
      row_idx, col_idx, g_node, segmax, y, denom, pooled, n_edges);

  // out = (pooled @ W_out^T) / (denom+1e-16) + b_out   (norm fused into epilogue)
  gemm_nt_wmma<true><<<(gemm_threads + 255) / 256, 256, 0, stream>>>(
      pooled, W_out, b_out, denom, out, n_nodes, D_FEAT, D_FEAT);
}